// SWGATLayer_10093173145807
// MI455X (gfx1250) — compile-verified
//
#include <hip/hip_runtime.h>
#include <hip/hip_bf16.h>

#define IN_DIM   256
#define OUT_DIM  128

typedef __attribute__((ext_vector_type(2))) float v2f;
typedef __attribute__((ext_vector_type(8))) float v8f;

// ---------- monotonic float <-> uint key (for atomicMax on floats) ----------
__device__ __forceinline__ unsigned fkey(float f) {
    unsigned b = __float_as_uint(f);
    return b ^ ((b & 0x80000000u) ? 0xFFFFFFFFu : 0x80000000u);
}
__device__ __forceinline__ float funkey(unsigned k) {
    unsigned b = k ^ ((k & 0x80000000u) ? 0x80000000u : 0xFFFFFFFFu);
    return __uint_as_float(b);
}

// ---------- 0: init out / max-key / denom ----------
__global__ __launch_bounds__(256) void init_kernel(unsigned* __restrict__ mkey,
                                                   float* __restrict__ denom,
                                                   float* __restrict__ out,
                                                   int nw, long long out_elems) {
    long long i = (long long)blockIdx.x * 256 + threadIdx.x;
    if (i < nw) { mkey[i] = fkey(-__builtin_huge_valf()); denom[i] = 0.f; }
    if (i < out_elems) out[i] = 0.f;
}

// ---------- 1: z = h @ W_fc.T via V_WMMA_F32_16X16X4_F32 ----------
// One wave -> one 16-row tile of z, all 128 output cols (8 accumulators).
__global__ __launch_bounds__(256) void gemm_z_kernel(const float* __restrict__ h,
                                                     const float* __restrict__ Wfc,
                                                     float* __restrict__ z,
                                                     int row_tiles) {
    int wave = blockIdx.x * 8 + (threadIdx.x >> 5);
    int lane = threadIdx.x & 31;
    if (wave >= row_tiles) return;               // whole-wave uniform: EXEC all-1s inside

    const int row0  = wave * 16;
    const int m     = lane & 15;                 // A: row within tile / B: col within tile
    const int khalf = (lane >> 4) * 2;           // K sub-offset per ISA A/B f32 layout

    const float* __restrict__ hrow = h + (size_t)(row0 + m) * IN_DIM + khalf;

    v8f acc[8];
#pragma unroll
    for (int t = 0; t < 8; ++t)
        acc[t] = (v8f){0.f, 0.f, 0.f, 0.f, 0.f, 0.f, 0.f, 0.f};

    for (int k0 = 0; k0 < IN_DIM; k0 += 4) {
        v2f a = *(const v2f*)(hrow + k0);        // A 16x4: lane<16 -> K=k0,k0+1 ; else K=k0+2,k0+3
#pragma unroll
        for (int nt = 0; nt < 8; ++nt) {
            // B[k][n] = Wfc[n][k]  (z = h @ Wfc^T)
            v2f b = *(const v2f*)(Wfc + (size_t)(nt * 16 + m) * IN_DIM + k0 + khalf);
            acc[nt] = __builtin_amdgcn_wmma_f32_16x16x4_f32(
                false, a, false, b, (short)0, acc[nt], false, false);
        }
    }

    // C/D layout: VGPR v, lane l -> row M = v + 8*(l>>4), col N = l&15
    const int mhi = (lane >> 4) * 8;
#pragma unroll
    for (int nt = 0; nt < 8; ++nt)
#pragma unroll
        for (int v = 0; v < 8; ++v)
            z[(size_t)(row0 + mhi + v) * OUT_DIM + nt * 16 + (lane & 15)] = acc[nt][v];
}

// ---------- 2: per-node attention partials  p = z·a1, q = z·a2 ----------
__global__ __launch_bounds__(256) void node_score_kernel(const float* __restrict__ z,
                                                         const float* __restrict__ Wattn,
                                                         float* __restrict__ p,
                                                         float* __restrict__ q,
                                                         int n_nodes) {
    int node = blockIdx.x * 8 + (threadIdx.x >> 5);
    int lane = threadIdx.x & 31;
    if (node >= n_nodes) return;
    float4 zv = ((const float4*)(z + (size_t)node * OUT_DIM))[lane];   // 32 lanes x float4 = 128
    float4 a1 = ((const float4*)Wattn)[lane];
    float4 a2 = ((const float4*)(Wattn + OUT_DIM))[lane];
    float ps = zv.x * a1.x + zv.y * a1.y + zv.z * a1.z + zv.w * a1.w;
    float qs = zv.x * a2.x + zv.y * a2.y + zv.z * a2.z + zv.w * a2.w;
#pragma unroll
    for (int off = 16; off > 0; off >>= 1) {
        ps += __shfl_down(ps, off, 32);
        qs += __shfl_down(qs, off, 32);
    }
    if (lane == 0) { p[node] = ps; q[node] = qs; }
}

// ---------- 3: edge scores + segment max ----------
__global__ __launch_bounds__(256) void edge_score_kernel(const int* __restrict__ src,
                                                         const int* __restrict__ dst,
                                                         const float* __restrict__ p,
                                                         const float* __restrict__ q,
                                                         float* __restrict__ escore,
                                                         unsigned* __restrict__ mkey,
                                                         int n_edges) {
    int e = blockIdx.x * 256 + threadIdx.x;
    if (e >= n_edges) return;
    float s = p[src[e]] + q[dst[e]];
    s = (s > 0.f) ? s : 0.01f * s;               // leaky_relu(0.01)
    if (s == 0.f) s = -1000.f;                   // DGL mailbox-padding mask
    escore[e] = s;
    atomicMax(mkey + dst[e], fkey(s));
}

// ---------- 4: exp + segment sum (denominator) ----------
__global__ __launch_bounds__(256) void edge_exp_kernel(const int* __restrict__ dst,
                                                       const float* __restrict__ escore,
                                                       const unsigned* __restrict__ mkey,
                                                       float* __restrict__ denom,
                                                       int n_edges) {
    int e = blockIdx.x * 256 + threadIdx.x;
    if (e >= n_edges) return;
    int d = dst[e];
    float ex = __expf(escore[e] - funkey(mkey[d]));
    unsafeAtomicAdd(denom + d, ex);              // global_atomic_add_f32
}

// ---------- 5: weighted aggregation: out[dst] += alpha * z[src] ----------
// 128 threads per edge, one output feature each; z gathers hit L2 (z = 51MB << 192MB).
__global__ __launch_bounds__(256) void aggregate_kernel(const int* __restrict__ src,
                                                        const int* __restrict__ dst,
                                                        const float* __restrict__ escore,
                                                        const unsigned* __restrict__ mkey,
                                                        const float* __restrict__ denom,
                                                        const float* __restrict__ z,
                                                        float* __restrict__ out,
                                                        int n_edges) {
    long long t = (long long)blockIdx.x * 256 + threadIdx.x;
    int e = (int)(t >> 7);
    int f = (int)(t & 127);
    if (e >= n_edges) return;
    int d = dst[e];
    float alpha = __expf(escore[e] - funkey(mkey[d])) / denom[d];
    float v = z[(size_t)src[e] * OUT_DIM + f];
    unsafeAtomicAdd(out + (size_t)d * OUT_DIM + f, alpha * v);
}

extern "C" void kernel_launch(void* const* d_in, const int* in_sizes, int n_in,
                              void* d_out, int out_size, void* d_ws, size_t ws_size,
                              hipStream_t stream) {
    const float* h     = (const float*)d_in[0];
    const int*   src   = (const int*)d_in[1];
    const int*   dst   = (const int*)d_in[2];
    const float* Wfc   = (const float*)d_in[3];
    const float* Wattn = (const float*)d_in[4];
    float* out = (float*)d_out;

    const int n_nodes = in_sizes[0] / IN_DIM;    // 100000
    const int n_edges = in_sizes[1];             // 500000
    const int nw      = out_size / OUT_DIM;      // 50000

    // workspace carve-up
    char* ws = (char*)d_ws;
    size_t off = 0;
    auto carve = [&](size_t bytes) -> void* {
        void* ptr = ws + off;
        off += (bytes + 255) & ~(size_t)255;
        return ptr;
    };
    float*    z      = (float*)carve(sizeof(float) * (size_t)n_nodes * OUT_DIM);
    float*    p      = (float*)carve(sizeof(float) * (size_t)n_nodes);
    float*    q      = (float*)carve(sizeof(float) * (size_t)n_nodes);
    float*    escore = (float*)carve(sizeof(float) * (size_t)n_edges);
    unsigned* mkey   = (unsigned*)carve(sizeof(unsigned) * (size_t)nw);
    float*    denom  = (float*)carve(sizeof(float) * (size_t)nw);
    (void)ws_size;

    // 0: init output + softmax state
    {
        long long out_elems = (long long)nw * OUT_DIM;
        int blocks = (int)((out_elems + 255) / 256);
        init_kernel<<<blocks, 256, 0, stream>>>(mkey, denom, out, nw, out_elems);
    }
    // 1: z = h @ Wfc^T  (fp32 WMMA)
    {
        int row_tiles = n_nodes / 16;            // 6250
        int blocks = (row_tiles + 7) / 8;
        gemm_z_kernel<<<blocks, 256, 0, stream>>>(h, Wfc, z, row_tiles);
    }
    // 2: per-node p/q
    {
        int blocks = (n_nodes + 7) / 8;
        node_score_kernel<<<blocks, 256, 0, stream>>>(z, Wattn, p, q, n_nodes);
    }
    // 3: edge scores + segment max
    {
        int blocks = (n_edges + 255) / 256;
        edge_score_kernel<<<blocks, 256, 0, stream>>>(src, dst, p, q, escore, mkey, n_edges);
    }
    // 4: exp + denom
    {
        int blocks = (n_edges + 255) / 256;
        edge_exp_kernel<<<blocks, 256, 0, stream>>>(dst, escore, mkey, denom, n_edges);
    }
    // 5: weighted scatter-add aggregation
    {
        long long threads = (long long)n_edges * OUT_DIM;
        int blocks = (int)((threads + 255) / 256);
        aggregate_kernel<<<blocks, 256, 0, stream>>>(src, dst, escore, mkey, denom, z, out, n_edges);
    }
}